// SparseLinear_16149077033887
// MI455X (gfx1250) — compile-verified
//
#include <hip/hip_runtime.h>
#include <cstdint>

// ---- problem constants (match reference) ----
#define NNZ_TOTAL   1048576
#define OUT_UNITS   128
#define N_ROWS_C    16384

// ---- kernel config ----
#define CHUNK       128                      // nnz per wave
#define WAVES_PB    8                        // waves per block (wave32)
#define BLOCK_T     (WAVES_PB * 32)          // 256 threads
#define NWAVES      (NNZ_TOTAL / CHUNK)      // 8192
#define NBLOCKS     (NWAVES / WAVES_PB)      // 1024

// CDNA5 async global->LDS copy (16B per lane), tracked by ASYNCcnt.
// GVS addressing: lds_byte_addr (VGPR), 32-bit byte offset (VGPR), 64-bit base (SGPR pair).
__device__ __forceinline__ void async_copy_b128(uint32_t lds_byte_addr,
                                                uint32_t voff,
                                                uint64_t sbase) {
  asm volatile("global_load_async_to_lds_b128 %0, %1, %2"
               :: "v"(lds_byte_addr), "v"(voff), "s"(sbase)
               : "memory");
}

__device__ __forceinline__ void wait_async0() {
#if __has_builtin(__builtin_amdgcn_s_wait_asynccnt)
  __builtin_amdgcn_s_wait_asynccnt(0);
#else
  asm volatile("s_wait_asynccnt 0x0" ::: "memory");
#endif
}

__global__ void __launch_bounds__(256)
SparseLinear_zero_kernel(float4* __restrict__ out, int n4) {
  int i = blockIdx.x * blockDim.x + threadIdx.x;
  int stride = gridDim.x * blockDim.x;
  for (; i < n4; i += stride) out[i] = make_float4(0.f, 0.f, 0.f, 0.f);
}

__global__ void __launch_bounds__(BLOCK_T)
SparseLinear_16149077033887_kernel(const float* __restrict__ values,
                                   const float* __restrict__ wmat,
                                   const long long* __restrict__ rows,
                                   const long long* __restrict__ cols,
                                   float* __restrict__ out) {
  __shared__ float     s_val[WAVES_PB][CHUNK];   //  4KB
  __shared__ long long s_row[WAVES_PB][CHUNK];   //  8KB
  __shared__ long long s_col[WAVES_PB][CHUNK];   //  8KB

  const uint32_t lane = threadIdx.x & 31u;
  const uint32_t wv   = threadIdx.x >> 5;
  const uint32_t gw   = blockIdx.x * WAVES_PB + wv;
  const uint32_t k0   = gw * CHUNK;

  // ---- async bulk stage of this wave's nnz metadata into LDS ----
  {
    uint64_t vb    = (uint64_t)(uintptr_t)(const void*)(values + k0);
    uint32_t lds_v = (uint32_t)(uintptr_t)&s_val[wv][0];
#pragma unroll
    for (uint32_t i = 0; i < (CHUNK * 4u) / (32u * 16u); ++i) {   // 1 inst
      uint32_t off = (i * 32u + lane) * 16u;
      async_copy_b128(lds_v + off, off, vb);
    }
    uint64_t rb    = (uint64_t)(uintptr_t)(const void*)(rows + k0);
    uint32_t lds_r = (uint32_t)(uintptr_t)&s_row[wv][0];
#pragma unroll
    for (uint32_t i = 0; i < (CHUNK * 8u) / (32u * 16u); ++i) {   // 2 insts
      uint32_t off = (i * 32u + lane) * 16u;
      async_copy_b128(lds_r + off, off, rb);
    }
    uint64_t cb    = (uint64_t)(uintptr_t)(const void*)(cols + k0);
    uint32_t lds_c = (uint32_t)(uintptr_t)&s_col[wv][0];
#pragma unroll
    for (uint32_t i = 0; i < (CHUNK * 8u) / (32u * 16u); ++i) {   // 2 insts
      uint32_t off = (i * 32u + lane) * 16u;
      async_copy_b128(lds_c + off, off, cb);
    }
  }
  wait_async0();

  // ---- accumulate: each lane owns 4 contiguous output columns ----
  const uint32_t cbase = lane * 4u;
  float ax = 0.f, ay = 0.f, az = 0.f, aw = 0.f;
  int cur = __builtin_amdgcn_readfirstlane((int)s_row[wv][0]);   // SGPR

#pragma unroll 8
  for (int k = 0; k < CHUNK; ++k) {
    // force wave-uniform values into SGPRs: SALU branch + saddr-form gather
    int r = __builtin_amdgcn_readfirstlane((int)s_row[wv][k]);
    if (r != cur) {
      float* o = out + (size_t)cur * OUT_UNITS + cbase;
      __hip_atomic_fetch_add(o + 0, ax, __ATOMIC_RELAXED, __HIP_MEMORY_SCOPE_AGENT);
      __hip_atomic_fetch_add(o + 1, ay, __ATOMIC_RELAXED, __HIP_MEMORY_SCOPE_AGENT);
      __hip_atomic_fetch_add(o + 2, az, __ATOMIC_RELAXED, __HIP_MEMORY_SCOPE_AGENT);
      __hip_atomic_fetch_add(o + 3, aw, __ATOMIC_RELAXED, __HIP_MEMORY_SCOPE_AGENT);
      ax = ay = az = aw = 0.f;
      cur = r;
    }
    int   c  = __builtin_amdgcn_readfirstlane((int)s_col[wv][k]);
    int   vi = __builtin_amdgcn_readfirstlane(__builtin_bit_cast(int, s_val[wv][k]));
    float v  = __builtin_bit_cast(float, vi);
    const float4 wr = *(const float4*)(wmat + (size_t)c * OUT_UNITS + cbase);
    ax = fmaf(v, wr.x, ax);
    ay = fmaf(v, wr.y, ay);
    az = fmaf(v, wr.z, az);
    aw = fmaf(v, wr.w, aw);
  }
  // final flush
  {
    float* o = out + (size_t)cur * OUT_UNITS + cbase;
    __hip_atomic_fetch_add(o + 0, ax, __ATOMIC_RELAXED, __HIP_MEMORY_SCOPE_AGENT);
    __hip_atomic_fetch_add(o + 1, ay, __ATOMIC_RELAXED, __HIP_MEMORY_SCOPE_AGENT);
    __hip_atomic_fetch_add(o + 2, az, __ATOMIC_RELAXED, __HIP_MEMORY_SCOPE_AGENT);
    __hip_atomic_fetch_add(o + 3, aw, __ATOMIC_RELAXED, __HIP_MEMORY_SCOPE_AGENT);
  }
}

extern "C" void kernel_launch(void* const* d_in, const int* in_sizes, int n_in,
                              void* d_out, int out_size, void* d_ws, size_t ws_size,
                              hipStream_t stream) {
  (void)in_sizes; (void)n_in; (void)d_ws; (void)ws_size;
  const float*     values = (const float*)d_in[0];
  const float*     wmat   = (const float*)d_in[1];
  const long long* rows   = (const long long*)d_in[2];
  const long long* cols   = (const long long*)d_in[3];
  float*           out    = (float*)d_out;

  // zero the output (harness poisons it), stream-ordered before accumulation
  int n4 = out_size / 4;   // 16384*128/4 float4s
  SparseLinear_zero_kernel<<<1024, 256, 0, stream>>>((float4*)out, n4);

  SparseLinear_16149077033887_kernel<<<NBLOCKS, BLOCK_T, 0, stream>>>(
      values, wmat, rows, cols, out);
}